// ITASoftmax_83588653515079
// MI455X (gfx1250) — compile-verified
//
#include <hip/hip_runtime.h>
#include <stdint.h>

// ITA integer softmax, MI455X (gfx1250), wave32.
// Shapes fixed by the reference: B*H = 16, S = 2048, K = 2048 -> 32768 rows of 2048 int32.

#define K_DIM        2048
#define GROUP_W      16
#define GROUPS       (K_DIM / GROUP_W)          // 128 groups per row
#define CH           4                          // groups staged per pipeline step
#define NSTAGE       (GROUPS / CH)              // 32 pipeline steps
#define WG_THREADS   128
#define ROWS_PER_WG  128
#define STAGE_ROW_B  (CH * GROUP_W * 4)         // 256 bytes per row per stage
#define ROW_STRIDE_B 272                        // 256 + 16B pad (LDS bank decorrelation)
#define BUF_BYTES    (ROWS_PER_WG * ROW_STRIDE_B)  // 34816 B per buffer
#define NBUF         4                          // quad buffer, prefetch 2 stages deep

__device__ __forceinline__ int imax(int a, int b) { return a > b ? a : b; }
__device__ __forceinline__ int imin31(int v) { return v < 31 ? v : 31; }

// CDNA5 async global->LDS copy (ASYNCcnt-tracked), GVS addressing:
// global addr = SGPR64 base + VGPR32 offset; VDST VGPR = LDS byte offset.
__device__ __forceinline__ void async_b128(unsigned lds_off, unsigned voff,
                                           unsigned long long sbase) {
  asm volatile("global_load_async_to_lds_b128 %0, %1, %2"
               :: "v"(lds_off), "v"(voff), "s"(sbase)
               : "memory");
}

template <int N>
__device__ __forceinline__ void wait_async() {
  asm volatile("s_wait_asynccnt %0" :: "n"(N) : "memory");
}

__global__ __launch_bounds__(WG_THREADS)
void ita_softmax_kernel(const int* __restrict__ x, float* __restrict__ out) {
  __shared__ unsigned char tile[NBUF * BUF_BYTES];
  __shared__ int sm_gmax[ROWS_PER_WG];
  __shared__ int sm_inv[ROWS_PER_WG];

  const int tid = threadIdx.x;
  const size_t row0 = (size_t)blockIdx.x * ROWS_PER_WG;
  const unsigned long long xbase =
      (unsigned long long)(uintptr_t)(x + row0 * K_DIM);
  const unsigned smem_base = (unsigned)(uintptr_t)&tile[0];  // low 32 bits = LDS offset

  // Stage CH groups (256 B) for each of the 128 rows; 16 lanes cooperate per row so
  // each async burst covers 256 B of contiguous global memory. Per-lane offsets are
  // stage-invariant; the stage advance lives in the uniform SGPR base.
  auto issue = [&](int stage, int buf) {
    const unsigned long long sbase = xbase + (unsigned long long)stage * STAGE_ROW_B;
    const unsigned lbase = smem_base + (unsigned)buf * BUF_BYTES;
#pragma unroll
    for (int q = 0; q < 16; ++q) {
      int u   = q * WG_THREADS + tid;   // 16B unit id, 16 units per row
      int row = u >> 4;
      int c   = u & 15;
      async_b128(lbase + (unsigned)(row * ROW_STRIDE_B + c * 16),
                 (unsigned)(row * (K_DIM * 4) + c * 16), sbase);
    }
  };

  issue(0, 0);
  issue(1, 1);

  int gmax = -128;
  int psum = 0;

#pragma unroll 4
  for (int s = 0; s < NSTAGE; ++s) {
    if (s + 2 < NSTAGE) {
      issue(s + 2, (s + 2) & (NBUF - 1));
      wait_async<32>();      // everything older than the 2 prefetched stages landed
    } else if (s + 1 < NSTAGE) {
      wait_async<16>();
    } else {
      wait_async<0>();
    }
    __syncthreads();          // all waves' loads for stage s visible; also fences the
                              // buffer that iteration s+1 will refill (2-deep pipeline)

    const int4* myrow =
        (const int4*)(tile + (s & (NBUF - 1)) * BUF_BYTES + tid * ROW_STRIDE_B);
#pragma unroll
    for (int g = 0; g < CH; ++g) {
      int4 a = myrow[g * 4 + 0];
      int4 b = myrow[g * 4 + 1];
      int4 c = myrow[g * 4 + 2];
      int4 d = myrow[g * 4 + 3];

      // group max (16 elements)
      int m = imax(imax(imax(imax(a.x, a.y), imax(a.z, a.w)),
                        imax(imax(b.x, b.y), imax(b.z, b.w))),
                   imax(imax(imax(c.x, c.y), imax(c.z, c.w)),
                        imax(imax(d.x, d.y), imax(d.z, d.w))));

      // online-max update (EPS_MAX == 1.0 -> shift_of is identity on ints)
      int diff = m - gmax;
      int ssh  = diff > 0 ? imin31(diff) : 0;
      gmax     = diff > 0 ? m : gmax;

      // exp_sum = sum(256 >> min(gmax - elem, 31))
      int es = 0;
      es += 256 >> imin31(gmax - a.x);  es += 256 >> imin31(gmax - a.y);
      es += 256 >> imin31(gmax - a.z);  es += 256 >> imin31(gmax - a.w);
      es += 256 >> imin31(gmax - b.x);  es += 256 >> imin31(gmax - b.y);
      es += 256 >> imin31(gmax - b.z);  es += 256 >> imin31(gmax - b.w);
      es += 256 >> imin31(gmax - c.x);  es += 256 >> imin31(gmax - c.y);
      es += 256 >> imin31(gmax - c.z);  es += 256 >> imin31(gmax - c.w);
      es += 256 >> imin31(gmax - d.x);  es += 256 >> imin31(gmax - d.y);
      es += 256 >> imin31(gmax - d.z);  es += 256 >> imin31(gmax - d.w);

      psum = (psum >> ssh) + es;
    }
  }

  // inv = floor(255*256 / psum); psum >= 256 guaranteed (global-max term contributes 256)
  sm_gmax[tid] = gmax;
  sm_inv[tid]  = 65280 / psum;
  __syncthreads();

  // Pass B: fully coalesced sweep; reads hit L2 (this WG's 1 MB region just streamed).
  const int4* xb4 = (const int4*)(x + row0 * K_DIM);
  float4*     ob4 = (float4*)(out + row0 * K_DIM);
#pragma unroll 1
  for (int r = 0; r < ROWS_PER_WG; ++r) {
    const int gm  = sm_gmax[r];
    const int inv = sm_inv[r];
    const int4* xr   = xb4 + (size_t)r * (K_DIM / 4);
    float4*     orow = ob4 + (size_t)r * (K_DIM / 4);
#pragma unroll
    for (int t = 0; t < K_DIM / 4 / WG_THREADS; ++t) {  // 4 iterations
      int i = t * WG_THREADS + tid;
      int4 v = xr[i];
      float4 f;
      f.x = (float)((inv >> imin31(gm - v.x)) & 255);
      f.y = (float)((inv >> imin31(gm - v.y)) & 255);
      f.z = (float)((inv >> imin31(gm - v.z)) & 255);
      f.w = (float)((inv >> imin31(gm - v.w)) & 255);
      orow[i] = f;
    }
  }
}

extern "C" void kernel_launch(void* const* d_in, const int* in_sizes, int n_in,
                              void* d_out, int out_size, void* d_ws, size_t ws_size,
                              hipStream_t stream) {
  (void)n_in; (void)d_ws; (void)ws_size; (void)out_size;
  const int* x = (const int*)d_in[0];
  float* out = (float*)d_out;
  int rows   = in_sizes[0] / K_DIM;          // 32768
  int blocks = rows / ROWS_PER_WG;           // 256
  ita_softmax_kernel<<<dim3(blocks), dim3(WG_THREADS), 0, stream>>>(x, out);
}